// Always_33182917329254
// MI455X (gfx1250) — compile-verified
//
#include <hip/hip_runtime.h>
#include <float.h>

// Sliding-window minimum, window W=256, padding='last'.
// van Herk / Gil-Werman: per aligned 256-segment prefix/suffix min scans,
// out[t] = min(suffix[t], prefix[t+255]).  O(1) amortized work per output.
//
// CDNA5 path: async global->LDS B128 copies (ASYNCcnt) for the tile,
// wave32 shuffle min-scans, vectorized LDS access.

#define SEG      256                 // window size == segment size
#define NOUT_PB  3840                // outputs per block (15 * 256)
#define TILE     (NOUT_PB + SEG)     // 4096 elements staged in LDS
#define NTHR     256                 // 8 waves (wave32)
#define NWAVE    (NTHR / 32)
#define NSEG     (TILE / SEG)        // 16 segments per tile

typedef int v4i __attribute__((ext_vector_type(4)));
typedef __attribute__((address_space(1))) v4i* g_v4i_ptr;   // global (AS1)
typedef __attribute__((address_space(3))) v4i* l_v4i_ptr;   // LDS (AS3)

__global__ __launch_bounds__(NTHR)
void slidemin256_kernel(const float* __restrict__ sig,
                        float* __restrict__ out,
                        int T) {
    __shared__ float s_raw[TILE];
    __shared__ float s_pre[TILE];
    __shared__ float s_suf[TILE];

    const int base = blockIdx.x * NOUT_PB;
    const int tid  = threadIdx.x;

    // ---------------- Stage tile (+ halo) into LDS ----------------
    if (base + TILE <= T) {
#if __has_builtin(__builtin_amdgcn_global_load_async_to_lds_b128)
        #pragma unroll
        for (int i = 0; i < TILE / (NTHR * 4); ++i) {      // 4 async B128 ops
            const int e = (tid + i * NTHR) * 4;            // float index
            __builtin_amdgcn_global_load_async_to_lds_b128(
                (g_v4i_ptr)(sig + base + e),
                (l_v4i_ptr)(&s_raw[e]),
                /*imm_offset=*/0, /*cpol=*/0);
        }
  #if __has_builtin(__builtin_amdgcn_s_wait_asynccnt)
        __builtin_amdgcn_s_wait_asynccnt(0);
  #else
        asm volatile("s_wait_asynccnt 0" ::: "memory");
  #endif
#else
        #pragma unroll
        for (int i = 0; i < TILE / (NTHR * 4); ++i) {
            const int e = (tid + i * NTHR) * 4;
            *(float4*)(&s_raw[e]) = *(const float4*)(sig + base + e);
        }
#endif
    } else {
        // Ragged final tile: clamp index to T-1 == padding='last'.
        for (int e = tid; e < TILE; e += NTHR) {
            int gi = base + e;
            if (gi > T - 1) gi = T - 1;
            s_raw[e] = sig[gi];
        }
    }
    __syncthreads();

    // ---------------- Per-segment prefix/suffix min scans ----------------
    const int lane = tid & 31;
    const int wave = tid >> 5;
    const float INFV = FLT_MAX;

    for (int s = wave; s < NSEG; s += NWAVE) {
        const int eb = s * SEG + lane * 8;        // this lane's 8 elements
        float v[8];
        {
            float4 a = *(const float4*)(&s_raw[eb]);
            float4 b = *(const float4*)(&s_raw[eb + 4]);
            v[0] = a.x; v[1] = a.y; v[2] = a.z; v[3] = a.w;
            v[4] = b.x; v[5] = b.y; v[6] = b.z; v[7] = b.w;
        }

        // serial prefix min inside the lane
        float p[8];
        p[0] = v[0];
        #pragma unroll
        for (int k = 1; k < 8; ++k) p[k] = fminf(p[k - 1], v[k]);

        // inclusive wave32 min-scan of lane totals (left -> right)
        float incl = p[7];
        #pragma unroll
        for (int off = 1; off < 32; off <<= 1) {
            float o = __shfl_up(incl, off, 32);
            if (lane >= off) incl = fminf(incl, o);
        }
        float exl = __shfl_up(incl, 1, 32);
        if (lane == 0) exl = INFV;

        {
            float4 w0, w1;
            w0.x = fminf(exl, p[0]); w0.y = fminf(exl, p[1]);
            w0.z = fminf(exl, p[2]); w0.w = fminf(exl, p[3]);
            w1.x = fminf(exl, p[4]); w1.y = fminf(exl, p[5]);
            w1.z = fminf(exl, p[6]); w1.w = fminf(exl, p[7]);
            *(float4*)(&s_pre[eb])     = w0;
            *(float4*)(&s_pre[eb + 4]) = w1;
        }

        // serial suffix min inside the lane
        float q[8];
        q[7] = v[7];
        #pragma unroll
        for (int k = 6; k >= 0; --k) q[k] = fminf(q[k + 1], v[k]);

        // inclusive wave32 min-scan (right -> left)
        float inclr = q[0];
        #pragma unroll
        for (int off = 1; off < 32; off <<= 1) {
            float o = __shfl_down(inclr, off, 32);
            if (lane + off < 32) inclr = fminf(inclr, o);
        }
        float exr = __shfl_down(inclr, 1, 32);
        if (lane == 31) exr = INFV;

        {
            float4 w0, w1;
            w0.x = fminf(exr, q[0]); w0.y = fminf(exr, q[1]);
            w0.z = fminf(exr, q[2]); w0.w = fminf(exr, q[3]);
            w1.x = fminf(exr, q[4]); w1.y = fminf(exr, q[5]);
            w1.z = fminf(exr, q[6]); w1.w = fminf(exr, q[7]);
            *(float4*)(&s_suf[eb])     = w0;
            *(float4*)(&s_suf[eb + 4]) = w1;
        }
    }
    __syncthreads();

    // ---------------- Combine: out[t] = min(suf[t], pre[t+255]) ----------------
    #pragma unroll
    for (int i = 0; i < NOUT_PB / NTHR; ++i) {   // 15 outputs per thread
        const int tl = tid + i * NTHR;
        const int gi = base + tl;
        if (gi < T) out[gi] = fminf(s_suf[tl], s_pre[tl + SEG - 1]);
    }
}

extern "C" void kernel_launch(void* const* d_in, const int* in_sizes, int n_in,
                              void* d_out, int out_size, void* d_ws, size_t ws_size,
                              hipStream_t stream) {
    (void)n_in; (void)d_ws; (void)ws_size; (void)out_size;
    const float* sig = (const float*)d_in[0];
    float* out = (float*)d_out;
    const int T = in_sizes[0];
    const int nblocks = (T + NOUT_PB - 1) / NOUT_PB;
    slidemin256_kernel<<<nblocks, NTHR, 0, stream>>>(sig, out, T);
}